// ChannelWiseAttention_22016002359943
// MI455X (gfx1250) — compile-verified
//
#include <hip/hip_runtime.h>
#include <hip/hip_bf16.h>
#include <math.h>

typedef __attribute__((ext_vector_type(2))) float v2f;
typedef __attribute__((ext_vector_type(8))) float v8f;

#define B_DIM   32
#define C_DIM   2048
#define D_DIM   512
#define K_SEL   614   // max(1, int(C * 0.3))

// ---------------------------------------------------------------------------
// Kernel 1: Q/K projection via V_WMMA_F32_16X16X4_F32.
// 8 waves per block, one 16-channel tile per wave. B operand comes from a
// branch-free LDS table [Wq | Wk | zeros]: lane N reads column N's K-pair with
// a single unconditional ds_load_b64 (lanes N>=2 stream the zero page), so the
// inner loop is global_load_b64 + ds_load_b64 + v_wmma with no exec dances.
// ---------------------------------------------------------------------------
__global__ __launch_bounds__(256) void qk_proj_wmma(
    const float* __restrict__ x,
    const float* __restrict__ Wq, const float* __restrict__ bq,
    const float* __restrict__ Wk, const float* __restrict__ bk,
    float* __restrict__ Qout, float* __restrict__ Kout)
{
    __shared__ __align__(16) float w_s[3 * D_DIM];   // [Wq | Wk | zeros]

    const int t = threadIdx.x;
    for (int i = t; i < D_DIM; i += 256) {
        w_s[i]             = Wq[i];
        w_s[D_DIM + i]     = Wk[i];
        w_s[2 * D_DIM + i] = 0.f;
    }
    __syncthreads();

    const int lane = t & 31;
    const int wave = t >> 5;
    const int tile = blockIdx.x * 8 + wave;          // 0 .. B*C/16-1
    const int tilesPerBatch = C_DIM / 16;            // 128
    const int b  = tile / tilesPerBatch;
    const int c0 = (tile % tilesPerBatch) * 16;

    // A-matrix (32-bit 16x4): lanes 0-15 -> K=d+0,d+1 of row M=lane,
    // lanes 16-31 -> K=d+2,d+3 of row M=lane-16. B mirrors with N=lane&15.
    const int half = lane >> 4;
    const int row  = lane & 15;
    const int col  = (row < 2) ? row : 2;            // 0=Wq, 1=Wk, 2=zero page

    const float* __restrict__ xrow =
        x + ((size_t)b * C_DIM + (size_t)(c0 + row)) * D_DIM + half * 2;
    const float* bptr = &w_s[col * D_DIM + half * 2];

    v8f acc = {0.f, 0.f, 0.f, 0.f, 0.f, 0.f, 0.f, 0.f};

    for (int d = 0; d < D_DIM; d += 4) {
        v2f a  = *(const v2f*)(xrow + d);            // global_load_b64
        v2f bv = *(const v2f*)(bptr + d);            // ds_load_b64 (uniform path)
        // 8 args: (neg_a, A, neg_b, B, c_mod, C, reuse_a, reuse_b)
        acc = __builtin_amdgcn_wmma_f32_16x16x4_f32(
            false, a, false, bv, (short)0, acc, false, false);
    }

    // C/D layout: VGPR j, lanes 0-15 -> D[M=j][N=lane]; lanes 16-31 -> D[M=j+8][N=lane-16].
    const float bqv = bq[0];
    const float bkv = bk[0];
    const size_t base = (size_t)b * C_DIM + c0;

    if (lane == 0)  { for (int j = 0; j < 8; ++j) Qout[base + j]     = acc[j] + bqv; }
    if (lane == 16) { for (int j = 0; j < 8; ++j) Qout[base + 8 + j] = acc[j] + bqv; }
    if (lane == 1)  { for (int j = 0; j < 8; ++j) Kout[base + j]     = acc[j] + bkv; }
    if (lane == 17) { for (int j = 0; j < 8; ++j) Kout[base + 8 + j] = acc[j] + bkv; }
}

// ---------------------------------------------------------------------------
// Kernel 2a: per-row softmax stats of the rank-1 score matrix.
// score[b,c,m] = Q[b,c]*K[b,m] (scale = 1 since KDIM=1), so
// M_c = q_c * (q_c>=0 ? max(K) : min(K)),  S_c = sum_m exp(q_c*k_m - M_c).
// Grid: B * C/256 blocks; one c per thread; K staged in LDS, float4 reads.
// ---------------------------------------------------------------------------
__global__ __launch_bounds__(256) void softmax_stats(
    const float* __restrict__ Q, const float* __restrict__ K,
    float* __restrict__ M_out, float* __restrict__ A_out)
{
    __shared__ __align__(16) float k_s[C_DIM];
    __shared__ float red_mx[256], red_mn[256];

    const int blocksPerBatch = C_DIM / 256;          // 8
    const int b     = blockIdx.x / blocksPerBatch;
    const int cBase = (blockIdx.x % blocksPerBatch) * 256;
    const int t     = threadIdx.x;

    const float* __restrict__ Kb = K + (size_t)b * C_DIM;
    for (int i = t; i < C_DIM; i += 256) k_s[i] = Kb[i];
    __syncthreads();

    float kmx = -INFINITY, kmn = INFINITY;
    for (int i = t; i < C_DIM; i += 256) {
        const float v = k_s[i];
        kmx = fmaxf(kmx, v);
        kmn = fminf(kmn, v);
    }
    red_mx[t] = kmx; red_mn[t] = kmn;
    __syncthreads();
    for (int s = 128; s > 0; s >>= 1) {
        if (t < s) {
            red_mx[t] = fmaxf(red_mx[t], red_mx[t + s]);
            red_mn[t] = fminf(red_mn[t], red_mn[t + s]);
        }
        __syncthreads();
    }
    kmx = red_mx[0]; kmn = red_mn[0];

    const int   c = cBase + t;
    const float q = Q[(size_t)b * C_DIM + c];
    const float M = (q >= 0.f) ? q * kmx : q * kmn;

    float s = 0.f;
    for (int m = 0; m < C_DIM; m += 4) {
        const float4 kv = *(const float4*)&k_s[m];   // ds_load_b128
        s += __expf(fmaf(q, kv.x, -M));
        s += __expf(fmaf(q, kv.y, -M));
        s += __expf(fmaf(q, kv.z, -M));
        s += __expf(fmaf(q, kv.w, -M));
    }
    M_out[(size_t)b * C_DIM + c] = M;
    A_out[(size_t)b * C_DIM + c] = 1.f / s;
}

// ---------------------------------------------------------------------------
// Kernel 2b: importance[b,m] = (1/C) * sum_c exp(q_c*k_m - M_c) * invS_c.
// Grid: B * C/256 blocks; one m per thread; q/M/invS staged in LDS.
// ---------------------------------------------------------------------------
__global__ __launch_bounds__(256) void importance_kernel(
    const float* __restrict__ Q, const float* __restrict__ K,
    const float* __restrict__ M_in, const float* __restrict__ A_in,
    float* __restrict__ imp)
{
    __shared__ __align__(16) float q_s[C_DIM];
    __shared__ __align__(16) float m_s[C_DIM];
    __shared__ __align__(16) float a_s[C_DIM];

    const int blocksPerBatch = C_DIM / 256;
    const int b     = blockIdx.x / blocksPerBatch;
    const int mBase = (blockIdx.x % blocksPerBatch) * 256;
    const int t     = threadIdx.x;
    const size_t base = (size_t)b * C_DIM;

    for (int i = t; i < C_DIM; i += 256) {
        q_s[i] = Q[base + i];
        m_s[i] = M_in[base + i];
        a_s[i] = A_in[base + i];
    }
    __syncthreads();

    const int   m  = mBase + t;
    const float km = K[base + m];

    float sum = 0.f;
    for (int c = 0; c < C_DIM; c += 4) {
        const float4 qv = *(const float4*)&q_s[c];
        const float4 Mv = *(const float4*)&m_s[c];
        const float4 av = *(const float4*)&a_s[c];
        sum += __expf(fmaf(qv.x, km, -Mv.x)) * av.x;
        sum += __expf(fmaf(qv.y, km, -Mv.y)) * av.y;
        sum += __expf(fmaf(qv.z, km, -Mv.z)) * av.z;
        sum += __expf(fmaf(qv.w, km, -Mv.w)) * av.w;
    }
    imp[base + m] = sum * (1.f / (float)C_DIM);
}

// ---------------------------------------------------------------------------
// Kernel 3: top-k by stable rank selection (descending, ties -> lower index),
// matching jax.lax.top_k ordering. rank < K_SEL => output slot == rank.
// Grid: B * C/256 blocks; one m per thread.
// ---------------------------------------------------------------------------
__global__ __launch_bounds__(256) void topk_kernel(
    const float* __restrict__ imp,
    int* __restrict__ idx_out, float* __restrict__ idx_f_out,
    float* __restrict__ k_out)
{
    __shared__ __align__(16) float v_s[C_DIM];

    const int blocksPerBatch = C_DIM / 256;
    const int b     = blockIdx.x / blocksPerBatch;
    const int mBase = (blockIdx.x % blocksPerBatch) * 256;
    const int t     = threadIdx.x;
    const size_t base = (size_t)b * C_DIM;

    for (int i = t; i < C_DIM; i += 256) v_s[i] = imp[base + i];
    __syncthreads();

    const int   m = mBase + t;
    const float v = v_s[m];
    int rank = 0;
    for (int c = 0; c < C_DIM; c += 4) {
        const float4 u = *(const float4*)&v_s[c];
        rank += ((u.x > v) || ((u.x == v) && (c + 0 < m))) ? 1 : 0;
        rank += ((u.y > v) || ((u.y == v) && (c + 1 < m))) ? 1 : 0;
        rank += ((u.z > v) || ((u.z == v) && (c + 2 < m))) ? 1 : 0;
        rank += ((u.w > v) || ((u.w == v) && (c + 3 < m))) ? 1 : 0;
    }
    if (rank < K_SEL) {
        idx_out[(size_t)b * K_SEL + rank]   = m;
        idx_f_out[(size_t)b * K_SEL + rank] = (float)m;
    }
    if (blockIdx.x == 0 && t == 0) k_out[0] = (float)K_SEL;
}

// ---------------------------------------------------------------------------
// Kernel 4: gather selected rows of x. 40 MB written; x (128 MB) fits the
// 192 MB L2 so the re-read mostly hits L2. One block per output row, float4.
// ---------------------------------------------------------------------------
__global__ __launch_bounds__(128) void gather_kernel(
    const float* __restrict__ x, const int* __restrict__ idx,
    float* __restrict__ out)
{
    const int row = blockIdx.x;            // b * K_SEL + r
    const int b   = row / K_SEL;
    const int ch  = idx[row];
    const float4* __restrict__ src =
        (const float4*)(x + ((size_t)b * C_DIM + (size_t)ch) * D_DIM);
    float4* __restrict__ dst = (float4*)(out + (size_t)row * D_DIM);
    for (int i = threadIdx.x; i < D_DIM / 4; i += 128)
        dst[i] = src[i];
}

// ---------------------------------------------------------------------------
extern "C" void kernel_launch(void* const* d_in, const int* in_sizes, int n_in,
                              void* d_out, int out_size, void* d_ws, size_t ws_size,
                              hipStream_t stream)
{
    const float* x  = (const float*)d_in[0];
    const float* Wq = (const float*)d_in[1];
    const float* bq = (const float*)d_in[2];
    const float* Wk = (const float*)d_in[3];
    const float* bk = (const float*)d_in[4];

    float* out       = (float*)d_out;
    float* feat_out  = out;                                    // B*K_SEL*D floats
    float* idx_f_out = out + (size_t)B_DIM * K_SEL * D_DIM;    // B*K_SEL floats
    float* k_out     = idx_f_out + (size_t)B_DIM * K_SEL;      // 1 float

    float* Qw   = (float*)d_ws;                                // B*C
    float* Kw   = Qw + (size_t)B_DIM * C_DIM;                  // B*C
    float* Mw   = Kw + (size_t)B_DIM * C_DIM;                  // B*C
    float* Aw   = Mw + (size_t)B_DIM * C_DIM;                  // B*C
    float* Iw   = Aw + (size_t)B_DIM * C_DIM;                  // B*C
    int*   idxw = (int*)(Iw + (size_t)B_DIM * C_DIM);          // B*K_SEL

    const int nTiles = B_DIM * (C_DIM / 16);                   // 4096
    qk_proj_wmma<<<nTiles / 8, 256, 0, stream>>>(x, Wq, bq, Wk, bk, Qw, Kw);

    const int nBlk = B_DIM * (C_DIM / 256);                    // 256
    softmax_stats<<<nBlk, 256, 0, stream>>>(Qw, Kw, Mw, Aw);
    importance_kernel<<<nBlk, 256, 0, stream>>>(Qw, Kw, Mw, Aw, Iw);
    topk_kernel<<<nBlk, 256, 0, stream>>>(Iw, idxw, idx_f_out, k_out);
    gather_kernel<<<B_DIM * K_SEL, 128, 0, stream>>>(x, idxw, feat_out);
}